// UnpillarNetworkScatter_85263690760898
// MI455X (gfx1250) — compile-verified
//
#include <hip/hip_runtime.h>

typedef __attribute__((ext_vector_type(2))) float v2f;
typedef __attribute__((ext_vector_type(8))) float v8f;

#define NPXY   (512 * 512)   // plane size (elements) of one channel
#define CCH    64            // channels in grid / point cloud
#define MAXP   100000        // points per batch

// One wave handles 16 points:
//   GEMM1: full(16x128) @ Yw^T(128x32) + Yb  via 32x2 V_WMMA_F32_16X16X4_F32
//   GEMM2: Y(16x32) @ Zw^T(32x3) + Zb       via LDS + VALU dot products
__global__ __launch_bounds__(256) void unpillar_fused_kernel(
    const float* __restrict__ grid,     // (B, 64, 512, 512)
    const float* __restrict__ pc,       // (B, P, 64)
    const long long* __restrict__ idx,  // (B, P, 64) int64, same value per channel
    const float* __restrict__ Yw,       // (32, 128)
    const float* __restrict__ Yb,       // (32)
    const float* __restrict__ Zw,       // (3, 32)
    const float* __restrict__ Zb,       // (3)
    float* __restrict__ out,            // (B, P, 3)
    int totalPoints)
{
    __shared__ float ldsY[8][16][33];   // per-wave 16x32 tile, padded stride

    const int tid  = threadIdx.x;
    const int wave = tid >> 5;
    const int lane = tid & 31;
    const int m    = lane & 15;         // point-in-tile (A rows / B,C cols)
    const int hi   = lane >> 4;         // K-half selector per WMMA layout

    if (blockIdx.x * 128 >= totalPoints) return;   // whole-block uniform guard

    const int gp0   = (blockIdx.x * 8 + wave) * 16;  // first flat point (b*P + p)
    const int flatP = gp0 + m;                       // this lane's point
    const int b     = flatP / MAXP;

    // pillar index for this lane's point (channel 0 of the broadcast int64 array)
    const long long pidx = idx[(long long)flatP * CCH];
    const float* gbase = grid + (size_t)b * CCH * NPXY + (size_t)pidx;
    const float* pbase = pc   + (size_t)flatP * CCH;

    // accumulators carry Y_b (bias depends only on output col N = 16t + m)
    v8f acc0, acc1;
    {
        const float b0 = Yb[m];
        const float b1 = Yb[16 + m];
        #pragma unroll
        for (int i = 0; i < 8; ++i) { acc0[i] = b0; acc1[i] = b1; }
    }

    // ---- K-steps 0..15: gathered grid embedding channels (strided planes) ----
    #pragma unroll
    for (int k = 0; k < 16; ++k) {
        const int f0 = 4 * k + 2 * hi;            // this lane's K pair
        v2f a;
        a.x = gbase[(size_t)f0 * NPXY];
        a.y = gbase[(size_t)(f0 + 1) * NPXY];
        v2f bv0 = *(const v2f*)(Yw + (size_t)m        * 128 + f0);
        v2f bv1 = *(const v2f*)(Yw + (size_t)(16 + m) * 128 + f0);
        acc0 = __builtin_amdgcn_wmma_f32_16x16x4_f32(false, a, false, bv0,
                                                     (short)0, acc0, false, false);
        acc1 = __builtin_amdgcn_wmma_f32_16x16x4_f32(false, a, false, bv1,
                                                     (short)0, acc1, false, false);
    }

    // ---- K-steps 16..31: point-cloud features (contiguous b64 loads) ----
    #pragma unroll
    for (int k = 16; k < 32; ++k) {
        const int f0 = 4 * k + 2 * hi;
        v2f a = *(const v2f*)(pbase + (f0 - CCH));
        v2f bv0 = *(const v2f*)(Yw + (size_t)m        * 128 + f0);
        v2f bv1 = *(const v2f*)(Yw + (size_t)(16 + m) * 128 + f0);
        acc0 = __builtin_amdgcn_wmma_f32_16x16x4_f32(false, a, false, bv0,
                                                     (short)0, acc0, false, false);
        acc1 = __builtin_amdgcn_wmma_f32_16x16x4_f32(false, a, false, bv1,
                                                     (short)0, acc1, false, false);
    }

    // ---- spill C layout to LDS: lane L, VGPR r -> Y[M = r + 8*hi][N] ----
    float* yrow = &ldsY[wave][0][0];
    #pragma unroll
    for (int r = 0; r < 8; ++r) {
        const int M = r + 8 * hi;
        yrow[M * 33 + m]      = acc0[r];
        yrow[M * 33 + 16 + m] = acc1[r];
    }
    __syncthreads();

    // ---- GEMM2: 48 outputs (16 points x 3) per wave, at most 2 per lane ----
    #pragma unroll
    for (int e = lane; e < 48; e += 32) {
        const int pt = e / 3;
        const int o  = e % 3;
        const float* yr = &ldsY[wave][pt][0];
        float acc = Zb[o];
        #pragma unroll
        for (int j = 0; j < 32; ++j)
            acc = fmaf(yr[j], Zw[o * 32 + j], acc);
        out[(long long)(gp0 + pt) * 3 + o] = acc;
    }
}

extern "C" void kernel_launch(void* const* d_in, const int* in_sizes, int n_in,
                              void* d_out, int out_size, void* d_ws, size_t ws_size,
                              hipStream_t stream) {
    const float*     grid = (const float*)d_in[0];
    const float*     pc   = (const float*)d_in[1];
    const long long* idx  = (const long long*)d_in[2];
    const float*     Yw   = (const float*)d_in[3];
    const float*     Yb   = (const float*)d_in[4];
    const float*     Zw   = (const float*)d_in[5];
    const float*     Zb   = (const float*)d_in[6];
    float*           out  = (float*)d_out;

    const int totalPoints = out_size / 3;               // B * MAX_POINTS = 400000
    const int blocks      = (totalPoints + 127) / 128;  // 128 points per block
    unpillar_fused_kernel<<<blocks, 256, 0, stream>>>(grid, pc, idx, Yw, Yb, Zw, Zb,
                                                      out, totalPoints);
}